// EmitLoss_84825604096252
// MI455X (gfx1250) — compile-verified
//
#include <hip/hip_runtime.h>

// CDNA5 / gfx1250. Wave32. One wave per batch, lane = sample (S == 32).
// Gram matrix G = M^T M (M is 32x8 per batch) via 8x V_WMMA_F32_16X16X4_F32.

typedef __attribute__((ext_vector_type(2))) float v2f;
typedef __attribute__((ext_vector_type(8))) float v8f;

#define KFAC 1.0297f
#define QL   0.1068f

// Constant-lane broadcast: v_readlane_b32 (no LDS round-trip like ds_bpermute).
__device__ __forceinline__ float rdlane(float v, int l) {
    return __int_as_float(__builtin_amdgcn_readlane(__float_as_int(v), l));
}

__device__ __forceinline__ void solve3_sym(float a00, float a01, float a02,
                                           float a11, float a12, float a22,
                                           float r0, float r1, float r2,
                                           float& p0, float& p1, float& p2) {
    // Cramer / adjugate for symmetric 3x3
    float C00 = a11 * a22 - a12 * a12;
    float C01 = a02 * a12 - a01 * a22;
    float C02 = a01 * a12 - a02 * a11;
    float C11 = a00 * a22 - a02 * a02;
    float C12 = a01 * a02 - a00 * a12;
    float C22 = a00 * a11 - a01 * a01;
    float det = a00 * C00 + a01 * C01 + a02 * C02;
    float inv = 1.0f / det;
    p0 = (C00 * r0 + C01 * r1 + C02 * r2) * inv;
    p1 = (C01 * r0 + C11 * r1 + C12 * r2) * inv;
    p2 = (C02 * r0 + C12 * r1 + C22 * r2) * inv;
}

__global__ __launch_bounds__(256) void emit_fit_kernel(
    const float* __restrict__ size2,   // [B, 32, 2]
    const float* __restrict__ field,   // [B, 32]
    const float* __restrict__ rmat,    // [2, 2, 2]
    float* __restrict__ out,           // [B, 8]
    int B)
{
    // Per-wave staging of M (8 components x 32 samples), component-major,
    // stride 33 floats so frag reads are bank-conflict-free.
    __shared__ float Mlds[8][8 * 33];

    const int lane = threadIdx.x & 31;
    const int wave = threadIdx.x >> 5;
    const int b    = blockIdx.x * (blockDim.x >> 5) + wave;
    if (b >= B) return;                       // wave-uniform: EXEC stays full

    float* myM = &Mlds[wave][0];

    // ---------------- per-sample physics (lane = sample) ----------------
    const int s = lane;
    float f   = field[(size_t)b * 32 + s];
    float kk  = fmaf(KFAC, f, 1e-8f);         // real part of complex k
    float w   = sqrtf(fabsf(kk));             // |sqrt(k)|
    float arg = w * QL;
    float sn, cs;
    __sincosf(arg, &sn, &cs);
    float e    = __expf(arg);
    float ei   = 1.0f / e;
    float ch   = 0.5f * (e + ei);
    float sh   = 0.5f * (e - ei);
    float winv = 1.0f / w;                    // w >= ~1e-4 (the +1e-8 shift)

    bool pos = (kk >= 0.0f);
    // k>=0: qx = [cos, sin/w, -w sin, cos], qy = [cosh, sinh/w, +w sinh, cosh]
    // k< 0: roles swap (complex sqrt branch): qx hyperbolic, qy trig.
    float qx00 = pos ? cs        : ch;
    float qx01 = pos ? sn * winv : sh * winv;
    float qx10 = pos ? -w * sn   : w * sh;
    float qy00 = pos ? ch        : cs;
    float qy01 = pos ? sh * winv : sn * winv;
    float qy10 = pos ? w * sh    : -w * sn;

    float r0x = rmat[0], r1x = rmat[1];       // r_mats[0] row 0 (uniform -> s_load)
    float r0y = rmat[4], r1y = rmat[5];       // r_mats[1] row 0

    float ax = fmaf(r0x, qx00, r1x * qx10);   // rm[0,0]
    float bx = fmaf(r0x, qx01, r1x * qx00);   // rm[0,1] (q11 == q00)
    float ay = fmaf(r0y, qy00, r1y * qy10);
    float by = fmaf(r0y, qy01, r1y * qy00);

    const float2 sz = *reinterpret_cast<const float2*>(size2 + ((size_t)b * 32 + s) * 2);

    // M row: [ax^2, 2ax*bx, bx^2, szx, ay^2, 2ay*by, by^2, szy]
    myM[0 * 33 + s] = ax * ax;
    myM[1 * 33 + s] = 2.0f * ax * bx;
    myM[2 * 33 + s] = bx * bx;
    myM[3 * 33 + s] = sz.x;
    myM[4 * 33 + s] = ay * ay;
    myM[5 * 33 + s] = 2.0f * ay * by;
    myM[6 * 33 + s] = by * by;
    myM[7 * 33 + s] = sz.y;
    // Same-wave DS ops are in-order (DScnt); no workgroup barrier needed.

    // ---------------- G = M^T M via WMMA f32 16x16x4, K-loop of 8 --------
    // 16x16x4 layout: A[m = L%16][k = 2*(L/16)+v], B[k = 2*(L/16)+v][n = L%16]
    // -> for A = M^T and B = M the fragment contents are IDENTICAL:
    //    frag[L][v] = M[sample = 4t + 2*(L>=16) + v][comp = L%16]
    //
    // Padding rows/cols 8..15: D[i][j] depends only on A-row i / B-col j, so
    // duplicated data there can never reach the 8x8 block we read. Clamp the
    // component index (m & 7) -> uniform control flow, LDS broadcast for the
    // duplicate lanes, DS loads batched ahead of the WMMA chain.
    const int mc    = (lane & 15) & 7;        // clamped component index
    const int khalf = (lane >> 4) << 1;       // 0 or 2
    const float* fbase = myM + mc * 33 + khalf;

    v2f fr[8];
    #pragma unroll
    for (int t = 0; t < 8; ++t) {
        fr[t].x = fbase[4 * t];
        fr[t].y = fbase[4 * t + 1];
    }

    v8f c = {};
    #pragma unroll
    for (int t = 0; t < 8; ++t) {
        // D = A*B + C ; accumulate over K blocks of 4 samples
        c = __builtin_amdgcn_wmma_f32_16x16x4_f32(
                /*neg_a=*/false, fr[t], /*neg_b=*/false, fr[t],
                /*c_mod=*/(short)0, c, /*reuse_a=*/false, /*reuse_b=*/false);
    }

    // G[i][j] lives in c[i] at lane j (rows 0..7, lanes 0..15 half).
    // x-plane: A = G[0..2][0..2], rhs = G[0..2][3]
    float a00x = rdlane(c[0], 0), a01x = rdlane(c[0], 1), a02x = rdlane(c[0], 2);
    float a11x = rdlane(c[1], 1), a12x = rdlane(c[1], 2), a22x = rdlane(c[2], 2);
    float rx0  = rdlane(c[0], 3), rx1  = rdlane(c[1], 3), rx2  = rdlane(c[2], 3);
    // y-plane: A = G[4..6][4..6], rhs = G[4..6][7]
    float a00y = rdlane(c[4], 4), a01y = rdlane(c[4], 5), a02y = rdlane(c[4], 6);
    float a11y = rdlane(c[5], 5), a12y = rdlane(c[5], 6), a22y = rdlane(c[6], 6);
    float ry0  = rdlane(c[4], 7), ry1  = rdlane(c[5], 7), ry2  = rdlane(c[6], 7);

    float px0, px1, px2, py0, py1, py2;
    solve3_sym(a00x, a01x, a02x, a11x, a12x, a22x, rx0, rx1, rx2, px0, px1, px2);
    solve3_sym(a00y, a01y, a02y, a11y, a12y, a22y, ry0, ry1, ry2, py0, py1, py2);
    float ex2 = px0 * px2 - px1 * px1;
    float ey2 = py0 * py2 - py1 * py1;

    if (lane == 0) {
        float4* o = reinterpret_cast<float4*>(out + (size_t)b * 8);
        o[0] = make_float4(px0, px1, px2, py0);
        o[1] = make_float4(py1, py2, ex2, ey2);
    }
}

extern "C" void kernel_launch(void* const* d_in, const int* in_sizes, int n_in,
                              void* d_out, int out_size, void* d_ws, size_t ws_size,
                              hipStream_t stream) {
    const float* size2 = (const float*)d_in[0];   // [B,32,2] f32
    const float* field = (const float*)d_in[1];   // [B,32]   f32
    const float* rmat  = (const float*)d_in[2];   // [2,2,2]  f32
    float* out = (float*)d_out;                   // [B,8]    f32

    int B = in_sizes[1] / 32;                     // field count / S
    const int wavesPerBlock = 8;                  // 256 threads
    int blocks = (B + wavesPerBlock - 1) / wavesPerBlock;
    emit_fit_kernel<<<blocks, 256, 0, stream>>>(size2, field, rmat, out, B);
}